// MultiBoxLoss_69020124447396
// MI455X (gfx1250) — compile-verified
//
#include <hip/hip_runtime.h>
#include <hip/hip_bf16.h>
#include <stdint.h>

typedef float v2f __attribute__((ext_vector_type(2)));
typedef float v8f __attribute__((ext_vector_type(8)));
typedef unsigned int  u32x4 __attribute__((ext_vector_type(4)));
typedef int  i32x4 __attribute__((ext_vector_type(4)));
typedef int  i32x8 __attribute__((ext_vector_type(8)));
typedef unsigned long long ull;

#define THRESH 0.5f
#define VAR0 0.1f
#define VAR1 0.2f
#define NEGPOS 3
#define CLS 81           // compile-time class count (matches reference)
#define TPB 256

// ---------------------------------------------------------------------------
// Block reduction (256 threads) using V_WMMA_F32_16X16X4_F32 as the reduce
// engine: D = A(16x4) x Ones(4x16) gives row-sums; one shfl_xor(16) finishes.
// Wave 0 only (EXEC all-1s satisfied: threads 0..31 are one full wave32).
// ---------------------------------------------------------------------------
__device__ __forceinline__ void reduce256_wmma(float v, float* red, float* gout) {
    const int t = threadIdx.x;
    red[t] = v;
    __syncthreads();
    if (t < 64) red[t] = red[t] + red[t + 64] + red[t + 128] + red[t + 192];
    __syncthreads();
    if (t < 32) {
        // 16x4 A layout: lanes 0-15 hold K=0,1 of row=lane; lanes 16-31 hold K=2,3.
        const int base = (t < 16) ? (t * 4) : ((t - 16) * 4 + 2);
        v2f A; A[0] = red[base]; A[1] = red[base + 1];
        v2f Bv; Bv[0] = 1.0f; Bv[1] = 1.0f;
        v8f Cz = {};
        v8f D = __builtin_amdgcn_wmma_f32_16x16x4_f32(false, A, false, Bv,
                                                      (short)0, Cz, false, false);
        float s = D[0] + D[1] + D[2] + D[3] + D[4] + D[5] + D[6] + D[7];
        s += __shfl_xor(s, 16, 32);
        if (t == 0) atomicAdd(gout, s);
    }
    __syncthreads();
}

// ---------------------------------------------------------------------------
// TDM: DMA a [rows x CLS] fp32 tile (row stride = CLS elements) into LDS.
// Must be called by a full single wave (e.g. threads 0..31); caller barriers.
// ---------------------------------------------------------------------------
__device__ __forceinline__ void tdm_load_tile(const float* gsrc, int rows,
                                              float* lds_dst) {
    unsigned lds_off = (unsigned)(uintptr_t)lds_dst;          // low 32b of flat = LDS offset
    unsigned long long ga = (unsigned long long)(uintptr_t)gsrc;

    u32x4 g0;
    g0[0] = 1u;                                               // count=1, user descriptor
    g0[1] = lds_off;                                          // lds_addr
    g0[2] = (unsigned)(ga & 0xFFFFFFFFull);                   // global_addr[31:0]
    g0[3] = (unsigned)((ga >> 32) & 0x01FFFFFFull) | (2u << 30); // global_addr[56:32] | type=2

    const unsigned dim0 = CLS;                                // tensor/tile width (elements)
    const unsigned dim1 = (unsigned)rows;
    i32x8 g1;
    g1[0] = (int)(2u << 16);                                  // data_size = 4 bytes
    g1[1] = (int)((dim0 & 0xFFFFu) << 16);                    // tensor_dim0[15:0]
    g1[2] = (int)((dim0 >> 16) | ((dim1 & 0xFFFFu) << 16));   // tensor_dim0[31:16], tensor_dim1[15:0]
    g1[3] = (int)((dim1 >> 16) | ((dim0 & 0xFFFFu) << 16));   // tensor_dim1[31:16], tile_dim0
    g1[4] = (int)(dim1 & 0xFFFFu);                            // tile_dim1 (tile_dim2 = 0)
    g1[5] = (int)dim0;                                        // tensor_dim0_stride[31:0]
    g1[6] = 0;
    g1[7] = 0;

    i32x4 gz = {0, 0, 0, 0};
#if __has_builtin(__builtin_amdgcn_tensor_load_to_lds)
  #if defined(__clang_major__) && (__clang_major__ >= 23)
    i32x8 gz8 = {0, 0, 0, 0, 0, 0, 0, 0};
    __builtin_amdgcn_tensor_load_to_lds(g0, g1, gz, gz, gz8, 0);
  #else
    __builtin_amdgcn_tensor_load_to_lds(g0, g1, gz, gz, 0);
  #endif
    __builtin_amdgcn_s_wait_tensorcnt(0);
#else
    // fallback: cooperative wave copy
    (void)g0; (void)g1; (void)gz;
    for (int i = (int)(threadIdx.x & 31); i < rows * CLS; i += 32)
        lds_dst[i] = gsrc[i];
#endif
}

// ---------------------------------------------------------------------------
// K0: zero accumulators / argmax keys
// ---------------------------------------------------------------------------
__global__ void k_init(ull* bestKey, int nKey, float* numPosB, int B,
                       float* accL, float* accCpos, float* accCneg) {
    int i = blockIdx.x * blockDim.x + threadIdx.x;
    if (i < nKey) bestKey[i] = 0ull;
    if (i < B) numPosB[i] = 0.0f;
    if (i == 0) { accL[0] = 0.0f; accCpos[0] = 0.0f; accCneg[0] = 0.0f; }
}

// ---------------------------------------------------------------------------
// K1: matching. Per prior: best truth (strict > == first-occurrence argmax).
// Per truth: best prior via u64 key (bits(ov)<<32)|(~p); only wave-max lanes
// touch the shared atomic (5 shfl_xor steps prune 32x of the atomic traffic).
// ---------------------------------------------------------------------------
__global__ void k_match(const float* __restrict__ priors,
                        const float* __restrict__ targets,
                        ull* __restrict__ bestKey,
                        float* __restrict__ ovBest, int* __restrict__ idxBest,
                        int P, int O) {
    const int b = blockIdx.y;
    const int p = blockIdx.x * blockDim.x + threadIdx.x;
    const bool valid = p < P;
    const int pc = valid ? p : (P - 1);
    __shared__ float tr[32 * 4];
    __shared__ ull blkKey[32];
    if (threadIdx.x < O * 4) {
        int o = threadIdx.x >> 2, j = threadIdx.x & 3;
        tr[threadIdx.x] = targets[((size_t)b * O + o) * 5 + j];
    }
    if (threadIdx.x < O) blkKey[threadIdx.x] = 0ull;
    __syncthreads();

    float cx = priors[pc * 4 + 0], cy = priors[pc * 4 + 1];
    float w  = priors[pc * 4 + 2], h  = priors[pc * 4 + 3];
    float ax1 = cx - 0.5f * w, ay1 = cy - 0.5f * h;
    float ax2 = cx + 0.5f * w, ay2 = cy + 0.5f * h;
    float areaP = (ax2 - ax1) * (ay2 - ay1);
    float best = -1.0f; int bi = 0;
    #pragma unroll 4
    for (int o = 0; o < O; ++o) {
        float tx1 = tr[o*4+0], ty1 = tr[o*4+1], tx2 = tr[o*4+2], ty2 = tr[o*4+3];
        float iw = fmaxf(fminf(ax2, tx2) - fmaxf(ax1, tx1), 0.0f);
        float ih = fmaxf(fminf(ay2, ty2) - fmaxf(ay1, ty1), 0.0f);
        float inter = iw * ih;
        float areaT = (tx2 - tx1) * (ty2 - ty1);
        float ov = inter / (areaP + areaT - inter);
        if (!valid) ov = -1.0f;
        if (ov > best) { best = ov; bi = o; }
        // wave-level max filter, then one shared u64 atomic from the winners
        float wmax = ov;
        #pragma unroll
        for (int d = 16; d > 0; d >>= 1) wmax = fmaxf(wmax, __shfl_xor(wmax, d, 32));
        if (valid && ov >= wmax) {
            ull key = ((ull)__float_as_uint(ov) << 32) |
                      (ull)(0xFFFFFFFFu - (unsigned)p);
            atomicMax(&blkKey[o], key);
        }
    }
    if (valid) {
        ovBest[(size_t)b * P + p] = best;
        idxBest[(size_t)b * P + p] = bi;
    }
    __syncthreads();
    if (threadIdx.x < O) atomicMax(&bestKey[b * O + threadIdx.x], blkKey[threadIdx.x]);
}

// ---------------------------------------------------------------------------
// K2: force-match fixup (sequential per batch: duplicate priors -> last wins)
// ---------------------------------------------------------------------------
__global__ void k_fixup(const ull* __restrict__ bestKey,
                        float* __restrict__ ovBest, int* __restrict__ idxBest,
                        int P, int O) {
    const int b = blockIdx.x;
    if (threadIdx.x == 0) {
        for (int o = 0; o < O; ++o) {
            ull k = bestKey[b * O + o];
            int p = (int)(0xFFFFFFFFu - (unsigned)(k & 0xFFFFFFFFull));
            ovBest[(size_t)b * P + p] = 2.0f;
            idxBest[(size_t)b * P + p] = o;
        }
    }
}

__device__ __forceinline__ float sl1(float d) {
    float ad = fabsf(d);
    return (ad < 1.0f) ? (0.5f * d * d) : (ad - 0.5f);
}

// ---------------------------------------------------------------------------
// K3: loc loss + logsumexp CE. The 255 MB conf stream is staged to LDS by the
// Tensor Data Mover (one DMA per block), then each thread scans its row from
// LDS (stride 81 floats = 17 mod 64 banks -> conflict-free). WMMA reductions.
// ---------------------------------------------------------------------------
__global__ void k_loss(const float* __restrict__ loc,
                       const float* __restrict__ conf,
                       const float* __restrict__ priors,
                       const float* __restrict__ targets,
                       const float* __restrict__ ovBest,
                       const int* __restrict__ idxBest,
                       float* __restrict__ ce, int* __restrict__ conft,
                       float* __restrict__ numPosB,
                       float* __restrict__ accL, float* __restrict__ accCpos,
                       int P, int O) {
    const int b = blockIdx.y;
    const int p0 = blockIdx.x * TPB;
    const int p = p0 + threadIdx.x;
    __shared__ float stage[TPB * CLS];    // 81 KB conf tile
    __shared__ float tr5[32 * 5];
    __shared__ float red[TPB];
    if (threadIdx.x < O * 5) tr5[threadIdx.x] = targets[(size_t)b * O * 5 + threadIdx.x];

    const int rows = min(TPB, P - p0);
    if (threadIdx.x < 32) {
        tdm_load_tile(conf + ((size_t)b * P + p0) * CLS, rows, stage);
    }
    __syncthreads();   // stage + tr5 visible to whole block

    float myL = 0.0f, myCpos = 0.0f, myPos = 0.0f;
    if (p < P) {
        const size_t bp = (size_t)b * P + p;
        float ov = ovBest[bp];
        int ti = idxBest[bp];
        int label = (int)tr5[ti * 5 + 4];
        int cf = (ov < THRESH) ? 0 : (label + 1);
        conft[bp] = cf;
        bool pos = cf > 0;

        if (pos) {
            float x1 = tr5[ti*5+0], y1 = tr5[ti*5+1], x2 = tr5[ti*5+2], y2 = tr5[ti*5+3];
            float pcx = priors[p*4+0], pcy = priors[p*4+1];
            float pw  = priors[p*4+2], ph  = priors[p*4+3];
            float gx = ((x1 + x2) * 0.5f - pcx) / (VAR0 * pw);
            float gy = ((y1 + y2) * 0.5f - pcy) / (VAR0 * ph);
            float gw = __logf((x2 - x1) / pw) / VAR1;
            float gh = __logf((y2 - y1) / ph) / VAR1;
            const float* lp = loc + bp * 4;
            myL = sl1(lp[0] - gx) + sl1(lp[1] - gy) + sl1(lp[2] - gw) + sl1(lp[3] - gh);
            myPos = 1.0f;
        }

        // online logsumexp over the LDS row
        const float* row = stage + threadIdx.x * CLS;
        float m = row[0], s = 1.0f;
        #pragma unroll 4
        for (int c = 1; c < CLS; ++c) {
            float x = row[c];
            if (x > m) { s = s * __expf(m - x) + 1.0f; m = x; }
            else       { s += __expf(x - m); }
        }
        float cev = (m + __logf(s)) - row[cf];
        ce[bp] = cev;
        if (pos) myCpos = cev;
    }

    reduce256_wmma(myL,    red, accL);
    reduce256_wmma(myCpos, red, accCpos);
    reduce256_wmma(myPos,  red, numPosB + b);
}

// ---------------------------------------------------------------------------
// K4: hard-negative mining via 4-round radix select on key = bits(lc) (lc>=0,
// bit pattern is order-preserving). Ties at tau share one value -> k_rem*tau.
// ---------------------------------------------------------------------------
__global__ void k_negmine(const float* __restrict__ ce,
                          const int* __restrict__ conft,
                          const float* __restrict__ numPosB,
                          float* __restrict__ accCneg, int P) {
    const int b = blockIdx.x, t = threadIdx.x;
    int np = (int)(numPosB[b] + 0.5f);
    long long nn = (long long)NEGPOS * np;
    long long cap = (long long)P - 1;
    int k = (int)(nn < cap ? nn : cap);
    if (k <= 0) return;   // uniform across the block (same b)

    const float* ceb = ce + (size_t)b * P;
    const int*   cfb = conft + (size_t)b * P;

    __shared__ unsigned int hist[256];
    __shared__ unsigned int sPrefix;
    __shared__ int sK;
    __shared__ float red[TPB];
    if (t == 0) { sPrefix = 0u; sK = k; }
    __syncthreads();

    for (int shift = 24; shift >= 0; shift -= 8) {
        hist[t] = 0u;
        __syncthreads();
        unsigned int prefix = sPrefix;
        unsigned int maskhi = (shift == 24) ? 0u : (0xFFFFFFFFu << (shift + 8));
        for (int p = t; p < P; p += TPB) {
            unsigned int key = (cfb[p] > 0) ? 0u : __float_as_uint(ceb[p]);
            if ((key & maskhi) == prefix)
                atomicAdd(&hist[(key >> shift) & 0xFFu], 1u);
        }
        __syncthreads();
        if (t == 0) {
            int kk = sK, digit = 0;
            for (int d = 255; d >= 0; --d) {
                unsigned int c = hist[d];
                if ((unsigned)kk > c) kk -= (int)c;
                else { digit = d; break; }
            }
            sK = kk;
            sPrefix = prefix | ((unsigned)digit << shift);
        }
        __syncthreads();
    }

    const unsigned int tau = sPrefix;
    const int krem = sK;
    float local = 0.0f;
    for (int p = t; p < P; p += TPB) {
        unsigned int key = (cfb[p] > 0) ? 0u : __float_as_uint(ceb[p]);
        if (key > tau) local += __uint_as_float(key);   // == ce (non-pos only)
    }
    if (t == 0) local += (float)krem * __uint_as_float(tau);

    reduce256_wmma(local, red, accCneg);
}

// ---------------------------------------------------------------------------
// K5: finalize -> d_out[0] = loss_l/N, d_out[1] = loss_c/N
// ---------------------------------------------------------------------------
__global__ void k_final(const float* accL, const float* accCpos,
                        const float* accCneg, const float* numPosB, int B,
                        float* out) {
    if (blockIdx.x == 0 && threadIdx.x == 0) {
        float tp = 0.0f;
        for (int i = 0; i < B; ++i) tp += numPosB[i];
        float N = tp + 1.0f;
        out[0] = accL[0] / N;
        out[1] = (accCpos[0] + accCneg[0]) / N;
    }
}

// ---------------------------------------------------------------------------
extern "C" void kernel_launch(void* const* d_in, const int* in_sizes, int n_in,
                              void* d_out, int out_size, void* d_ws, size_t ws_size,
                              hipStream_t stream) {
    const float* loc     = (const float*)d_in[0];
    const float* conf    = (const float*)d_in[1];
    const float* priors  = (const float*)d_in[2];
    const float* targets = (const float*)d_in[3];

    const int P  = in_sizes[2] / 4;
    const int BP = in_sizes[0] / 4;
    const int B  = BP / P;
    const int O  = in_sizes[3] / (B * 5);

    char* w = (char*)d_ws;
    size_t off = 0;
    auto take = [&](size_t bytes) -> void* {
        void* r = (void*)(w + off);
        off = (off + bytes + 255) & ~(size_t)255;
        return r;
    };
    ull*   bestKey = (ull*)  take((size_t)B * O * sizeof(ull));
    float* ovBest  = (float*)take((size_t)BP * sizeof(float));
    int*   idxBest = (int*)  take((size_t)BP * sizeof(int));
    float* ce      = (float*)take((size_t)BP * sizeof(float));
    int*   conft   = (int*)  take((size_t)BP * sizeof(int));
    float* numPosB = (float*)take((size_t)B * sizeof(float));
    float* accL    = (float*)take(sizeof(float));
    float* accCpos = (float*)take(sizeof(float));
    float* accCneg = (float*)take(sizeof(float));

    float* out = (float*)d_out;

    dim3 gridBP((P + TPB - 1) / TPB, B);

    k_init<<<(B * O + TPB - 1) / TPB, TPB, 0, stream>>>(
        bestKey, B * O, numPosB, B, accL, accCpos, accCneg);

    k_match<<<gridBP, TPB, 0, stream>>>(priors, targets, bestKey, ovBest, idxBest, P, O);

    k_fixup<<<B, 32, 0, stream>>>(bestKey, ovBest, idxBest, P, O);

    k_loss<<<gridBP, TPB, 0, stream>>>(loc, conf, priors, targets, ovBest, idxBest,
                                       ce, conft, numPosB, accL, accCpos, P, O);

    k_negmine<<<B, TPB, 0, stream>>>(ce, conft, numPosB, accCneg, P);

    k_final<<<1, 32, 0, stream>>>(accL, accCpos, accCneg, numPosB, B, out);
}